// MoE_36601711297094
// MI455X (gfx1250) — compile-verified
//
#include <hip/hip_runtime.h>

// MoE: E=64 routed experts (top-6) + 1 shared expert, H=2048, F=1408, T=512.
// HBM-bound on 2.2GB fp32 weights (~95us @ 23.3TB/s) -> full-precision
// V_WMMA_F32_16X16X4_F32; weight panels streamed into LDS with
// global_load_async_to_lds_b128 (phase 1) and the Tensor Data Mover (phase 2),
// double-buffered against s_wait_asynccnt / s_wait_tensorcnt.

#define HID  2048
#define FEXP 1408
#define NEXP 64
#define TOPK 6
#define NTOK 512
#define KS   32            // K rows per staged slab
#define BPAD 136           // staged panel row stride in floats (bank-clean)

typedef float v2f __attribute__((ext_vector_type(2)));
typedef float v8f __attribute__((ext_vector_type(8)));
typedef unsigned int u32;
typedef unsigned long long u64;
typedef u32 u32x4 __attribute__((ext_vector_type(4)));
typedef u32 u32x8 __attribute__((ext_vector_type(8)));

// -------------------------------------------------------------------------
__device__ __forceinline__ u32 lds_addr_of(const void* p) {
    // flat-address low 32 bits are the LDS byte offset (aperture mapping)
    return (u32)(uintptr_t)p;
}

__device__ __forceinline__ void async_copy_b128(u32 lds, const float* g) {
    asm volatile("global_load_async_to_lds_b128 %0, %1, off"
                 :: "v"(lds), "v"((u64)(uintptr_t)g) : "memory");
}

// TDM 2D tile load: rows x cols fp32 tile (cols contiguous, row stride in
// elements). D# pad: interval 128 dwords (code 6), amount 8 dwords (code 7)
// -> LDS row stride 136 floats == BPAD.
__device__ __forceinline__ void tdm_load_2d(u32 lds, const float* gp,
                                            u32 rows, u32 cols, u64 stride) {
    const u64 ga = (u64)(uintptr_t)gp;
    u32x4 g0;
    g0.s0 = 0x1u;                                            // count=1 valid
    g0.s1 = lds;                                             // lds_addr
    g0.s2 = (u32)ga;                                         // global_addr lo
    g0.s3 = (u32)((ga >> 32) & 0x01FFFFFFu) | (2u << 30);    // addr hi | type=2
    u32x8 g1;
    g1.s0 = 0x0F920000u;        // data_size=4B | pad_en | interval=128dw | amt=8dw
    g1.s1 = (cols & 0xFFFFu) << 16;                          // tensor_dim0 lo
    g1.s2 = (cols >> 16) | ((rows & 0xFFFFu) << 16);         // dim0 hi | dim1 lo
    g1.s3 = (rows >> 16) | ((cols & 0xFFFFu) << 16);         // dim1 hi | tile_dim0
    g1.s4 = (rows & 0xFFFFu);                                // tile_dim1 (dim2=0)
    g1.s5 = (u32)(stride & 0xFFFFFFFFu);                     // dim0_stride lo
    g1.s6 = (u32)((stride >> 32) & 0xFFFFu);                 // dim0_stride hi
    g1.s7 = 0u;
    u32x4 gz = {0u, 0u, 0u, 0u};                             // unused dims 2..4
    asm volatile("tensor_load_to_lds %0, %1, %2, %3"
                 :: "s"(g0), "s"(g1), "s"(gz), "s"(gz) : "memory");
}

// -------------------------------------------------------------------------
__global__ __launch_bounds__(256) void moe_zero_kernel(float* __restrict__ out, int n) {
    int i = blockIdx.x * blockDim.x + threadIdx.x;
    if (i < n) out[i] = 0.0f;
}

// -------------------------------------------------------------------------
__global__ __launch_bounds__(64) void moe_router_kernel(
    const float* __restrict__ x, const float* __restrict__ wr,
    float* __restrict__ combine)
{
    const int t = blockIdx.x;
    const int e = threadIdx.x;                       // 0..63
    const float* xr  = x  + (size_t)t * HID;
    const float* wrr = wr + (size_t)e * HID;
    float acc = 0.0f;
    for (int h = 0; h < HID; ++h) acc = __builtin_fmaf(xr[h], wrr[h], acc);

    __shared__ float lg[NEXP];
    lg[e] = acc;
    combine[t * NEXP + e] = 0.0f;
    __syncthreads();
    if (e == 0) {
        float mx = lg[0];
        for (int i = 1; i < NEXP; ++i) mx = fmaxf(mx, lg[i]);
        float s = 0.0f;
        for (int i = 0; i < NEXP; ++i) s += __expf(lg[i] - mx);
        const float inv = 1.0f / s;
        for (int k = 0; k < TOPK; ++k) {
            int bi = 0; float bv = -1e30f;
            for (int i = 0; i < NEXP; ++i)
                if (lg[i] > bv) { bv = lg[i]; bi = i; }
            combine[t * NEXP + bi] = __expf(bv - mx) * inv;
            lg[bi] = -1e30f;
        }
    }
}

// -------------------------------------------------------------------------
// grid = (32 token tiles, 65 experts); blockIdx.y==64 -> shared expert.
// 256 threads = 8 waves. Wave w handles column tile j*8+w -> the block
// consumes one contiguous 128-column weight panel per j (stageable 2D tile).
__global__ __launch_bounds__(256) void moe_expert_kernel(
    const float* __restrict__ x,
    const float* __restrict__ wg, const float* __restrict__ wv,
    const float* __restrict__ wo,
    const float* __restrict__ swg, const float* __restrict__ swv,
    const float* __restrict__ swo,
    const float* __restrict__ combine,
    float* __restrict__ out)
{
    __shared__ __align__(16) float Hs[16][FEXP + 4];            // 90.4 KB
    __shared__ __align__(16) float Bstage[2][2][KS][BPAD];      // 69.6 KB
    __shared__ float cwS[16];

    const int m    = blockIdx.x;
    const int e    = blockIdx.y;
    const int tid  = threadIdx.x;
    const int wid  = tid >> 5;
    const int lane = tid & 31;
    const int half = lane >> 4;
    const int l16  = lane & 15;
    const int col  = wid * 16 + l16;        // panel-local column of this lane

    if (tid < 16)
        cwS[tid] = (e < NEXP) ? combine[(m * 16 + tid) * NEXP + e] : 1.0f;
    __syncthreads();

    float csum = 0.0f;
    for (int i = 0; i < 16; ++i) csum += __builtin_fabsf(cwS[i]);
    if (csum == 0.0f) return;               // uniform: tile not routed here

    const float* gmat = (e < NEXP) ? wg + (size_t)e * HID * FEXP : swg;
    const float* vmat = (e < NEXP) ? wv + (size_t)e * HID * FEXP : swv;
    const float* omat = (e < NEXP) ? wo + (size_t)e * FEXP * HID : swo;

    // A-fragment: lane l16 owns row M=l16; lanes 0-15 K pair (k,k+1),
    // lanes 16-31 (k+2,k+3) -> single b64 per WMMA.
    const float* xa = x + (size_t)(m * 16 + l16) * HID + 2 * half;

    // ============ Phase 1: Hs = silu(X@Wg) * (X@Wv) ============
    for (int j = 0; j < FEXP / 128; ++j) {                 // 11 panels
        v8f accg = {};
        v8f accv = {};
        // stage slab s of both matrices: 64 row-copies, 8 per wave
        auto stage1 = [&](int s, int b) {
            #pragma unroll
            for (int q = 0; q < 8; ++q) {
                const int idx = wid * 8 + q;
                const int mat = idx >> 5;
                const int r   = idx & (KS - 1);
                const float* src = (mat ? vmat : gmat)
                                 + (u64)(s * KS + r) * FEXP + j * 128 + lane * 4;
                async_copy_b128(lds_addr_of(&Bstage[b][mat][r][0]) + lane * 16u, src);
            }
        };
        stage1(0, 0);
        #pragma unroll 1
        for (int s = 0; s < HID / KS; ++s) {               // 64 slabs
            if (s + 1 < HID / KS) {
                stage1(s + 1, (s + 1) & 1);
                asm volatile("s_wait_asynccnt 0x8" ::: "memory");
            } else {
                asm volatile("s_wait_asynccnt 0x0" ::: "memory");
            }
            __syncthreads();
            const float (*Bg)[BPAD] = Bstage[s & 1][0];
            const float (*Bv)[BPAD] = Bstage[s & 1][1];
            const float* ak = xa + s * KS;
            #pragma unroll
            for (int kk = 0; kk < KS; kk += 4) {
                v2f a = *(const v2f*)(ak + kk);
                const int r0 = kk + 2 * half;
                v2f bg, bv;
                bg.x = Bg[r0][col];  bg.y = Bg[r0 + 1][col];
                bv.x = Bv[r0][col];  bv.y = Bv[r0 + 1][col];
                accg = __builtin_amdgcn_wmma_f32_16x16x4_f32(
                    false, a, false, bg, (short)0, accg, false, false);
                accv = __builtin_amdgcn_wmma_f32_16x16x4_f32(
                    false, a, false, bv, (short)0, accv, false, false);
            }
            __syncthreads();
        }
        const int n0 = j * 128 + wid * 16;
        #pragma unroll
        for (int i = 0; i < 8; ++i) {                      // D: VGPR i -> M=i+8*half
            const float gg  = accg[i];
            const float sig = __builtin_amdgcn_rcpf(1.0f + __expf(-gg));
            Hs[i + 8 * half][n0 + l16] = gg * sig * accv[i];
        }
    }
    __syncthreads();

    // ============ Phase 2: out += cw * (Hs @ Wo), TDM-staged ============
    const float* hp = &Hs[l16][2 * half];
    #pragma unroll 1
    for (int j2 = 0; j2 < HID / 128; ++j2) {               // 16 panels
        v8f acc = {};
        if (tid == 0)                                       // TDM ignores EXEC
            tdm_load_2d(lds_addr_of(&Bstage[0][0][0][0]),
                        omat + (u64)j2 * 128, KS, 128, HID);
        #pragma unroll 1
        for (int s = 0; s < FEXP / KS; ++s) {              // 44 slabs
            if (s + 1 < FEXP / KS) {
                if (tid == 0)
                    tdm_load_2d(lds_addr_of(&Bstage[(s + 1) & 1][0][0][0]),
                                omat + (u64)((s + 1) * KS) * HID + j2 * 128,
                                KS, 128, HID);
                __builtin_amdgcn_s_wait_tensorcnt(1);
            } else {
                __builtin_amdgcn_s_wait_tensorcnt(0);
            }
            __syncthreads();
            const float (*Bo)[BPAD] = Bstage[s & 1][0];
            const float* ak = hp + s * KS;
            #pragma unroll
            for (int kk = 0; kk < KS; kk += 4) {
                v2f a = *(const v2f*)(ak + kk);            // ds_load_b64
                const int r0 = kk + 2 * half;
                v2f b;
                b.x = Bo[r0][col];  b.y = Bo[r0 + 1][col];
                acc = __builtin_amdgcn_wmma_f32_16x16x4_f32(
                    false, a, false, b, (short)0, acc, false, false);
            }
            __syncthreads();
        }
        const int n0 = j2 * 128 + wid * 16;
        #pragma unroll
        for (int i = 0; i < 8; ++i) {
            const int row = i + 8 * half;
            atomicAdd(&out[(size_t)(m * 16 + row) * HID + n0 + l16],
                      acc[i] * cwS[row]);
        }
    }
}

// -------------------------------------------------------------------------
extern "C" void kernel_launch(void* const* d_in, const int* in_sizes, int n_in,
                              void* d_out, int out_size, void* d_ws, size_t ws_size,
                              hipStream_t stream) {
    const float* x   = (const float*)d_in[0];
    const float* wr  = (const float*)d_in[1];
    const float* wg  = (const float*)d_in[2];
    const float* wv  = (const float*)d_in[3];
    const float* wo  = (const float*)d_in[4];
    const float* swg = (const float*)d_in[5];
    const float* swv = (const float*)d_in[6];
    const float* swo = (const float*)d_in[7];
    float* out = (float*)d_out;
    float* combine = (float*)d_ws;          // [512, 64] fp32

    const int n_out = NTOK * HID;
    moe_zero_kernel<<<(n_out + 255) / 256, 256, 0, stream>>>(out, n_out);
    moe_router_kernel<<<NTOK, 64, 0, stream>>>(x, wr, combine);

    dim3 grid(NTOK / 16, NEXP + 1);
    moe_expert_kernel<<<grid, 256, 0, stream>>>(x, wg, wv, wo, swg, swv, swo,
                                                combine, out);
}